// QuadPool2d_24893630447776
// MI455X (gfx1250) — compile-verified
//
#include <hip/hip_runtime.h>

// QuadPool gather: out[i] = weight[slot(pts[i])] * x[i] + bias[slot(pts[i])]
// slot = (ix*1024 + iy) & 0xFFFFF with ix/iy = clip(floor(p*1024), 0, 1023)
//
// MI455X strategy: pure HBM-bandwidth kernel (~400 MB traffic, ~17 us floor
// at 23.3 TB/s). Stream pts/x/out with non-temporal hints so the 8 MB
// weight+bias tables stay resident in the 192 MB L2 for the random gathers.
// 128-bit vector memory ops for all streaming traffic; gfx1250
// global_prefetch_b8 one grid-stride ahead.

typedef float v4f __attribute__((ext_vector_type(4)));

#define TILE_N      1024        // 2^DEPTH cells per axis
#define SLOT_MASK   0xFFFFFu    // KERNEL_SIZE - 1 (KERNEL_SIZE = 1<<20)

__device__ __forceinline__ int tile_coord(float p) {
    // clip(floor(p * 1024), 0, 1023)
    float f = floorf(p * (float)TILE_N);
    f = fminf(fmaxf(f, 0.0f), (float)(TILE_N - 1));
    return (int)f;
}

__device__ __forceinline__ unsigned slot_of(float px, float py) {
    return (unsigned)((tile_coord(px) << 10) | tile_coord(py)) & SLOT_MASK;
}

__global__ void __launch_bounds__(256)
quadpool_gather_kernel(const float* __restrict__ pts,     // [N,2]
                       const float* __restrict__ xv,      // [N]
                       const float* __restrict__ weight,  // [1<<20]
                       const float* __restrict__ bias,    // [1<<20]
                       float* __restrict__ out,           // [N]
                       int nvec,                          // N/4 four-point groups
                       int n)                             // total points
{
    const int stride = (int)(gridDim.x * blockDim.x);
    const int tid    = (int)(blockIdx.x * blockDim.x + threadIdx.x);

    const v4f* __restrict__ pts4 = (const v4f*)pts;  // 2 points per v4f
    const v4f* __restrict__ x4   = (const v4f*)xv;
    v4f*       __restrict__ out4 = (v4f*)out;

    for (int i = tid; i < nvec; i += stride) {
        // Prefetch next grid-stride iteration's streaming lines (gfx1250
        // global_prefetch_b8; speculative — OOB addresses are dropped).
        {
            const long long ni = (long long)i + (long long)stride;
            __builtin_prefetch((const void*)(pts + ni * 8), 0, 0);
            __builtin_prefetch((const void*)(xv  + ni * 4), 0, 0);
        }

        // Non-temporal 128-bit streaming loads: don't let 268 MB of stream
        // traffic evict the 8 MB gather tables from L2.
        v4f p01 = __builtin_nontemporal_load(&pts4[2 * (long long)i + 0]);
        v4f p23 = __builtin_nontemporal_load(&pts4[2 * (long long)i + 1]);
        v4f xx  = __builtin_nontemporal_load(&x4[i]);

        const unsigned s0 = slot_of(p01.x, p01.y);
        const unsigned s1 = slot_of(p01.z, p01.w);
        const unsigned s2 = slot_of(p23.x, p23.y);
        const unsigned s3 = slot_of(p23.z, p23.w);

        // Random 4 B gathers — expected to hit L2 (tables resident).
        const float w0 = weight[s0], b0 = bias[s0];
        const float w1 = weight[s1], b1 = bias[s1];
        const float w2 = weight[s2], b2 = bias[s2];
        const float w3 = weight[s3], b3 = bias[s3];

        v4f r;
        r.x = fmaf(w0, xx.x, b0);
        r.y = fmaf(w1, xx.y, b1);
        r.z = fmaf(w2, xx.z, b2);
        r.w = fmaf(w3, xx.w, b3);

        __builtin_nontemporal_store(r, &out4[i]);
    }

    // Scalar tail (n not divisible by 4): never taken for N = 16,777,216,
    // but kept for generality.
    for (int j = nvec * 4 + tid; j < n; j += stride) {
        const unsigned s = slot_of(pts[2 * (long long)j], pts[2 * (long long)j + 1]);
        out[j] = fmaf(weight[s], xv[j], bias[s]);
    }
}

extern "C" void kernel_launch(void* const* d_in, const int* in_sizes, int n_in,
                              void* d_out, int out_size, void* d_ws, size_t ws_size,
                              hipStream_t stream) {
    const float* pts    = (const float*)d_in[0];  // [N,2] float32
    const float* xv     = (const float*)d_in[1];  // [N]   float32
    const float* weight = (const float*)d_in[2];  // [1<<20] float32
    const float* bias   = (const float*)d_in[3];  // [1<<20] float32
    float*       out    = (float*)d_out;

    const int n    = in_sizes[1];   // number of points (x element count)
    const int nvec = n >> 2;        // four-point groups

    // 256 threads = 8 wave32 waves per block. Cap blocks so each thread
    // loops ~4x, making the one-stride-ahead prefetch meaningful while
    // still launching ~1M threads (32K waves) for latency hiding.
    const int threads   = 256;
    long long wanted    = ((long long)nvec + threads - 1) / threads;
    int blocks          = (int)(wanted < 4096 ? (wanted > 0 ? wanted : 1) : 4096);

    quadpool_gather_kernel<<<blocks, threads, 0, stream>>>(
        pts, xv, weight, bias, out, nvec, n);
}